// GraphPool_75746043232298
// MI455X (gfx1250) — compile-verified
//
#include <hip/hip_runtime.h>
#include <hip/hip_bf16.h>

#define N_NODES   100000
#define N_EDGES   3200000
#define IN_F      384
#define HID       64
#define NEG_SLOPE 0.01f

typedef __attribute__((ext_vector_type(2))) float v2f;
typedef __attribute__((ext_vector_type(8))) float v8f;

// ---------------- degree / norm ----------------

__global__ void init_deg_kernel(float* __restrict__ deg) {
    int i = blockIdx.x * blockDim.x + threadIdx.x;
    if (i < N_NODES) deg[i] = 1.0f;   // self-loop weight
}

__global__ void deg_accum_kernel(const int* __restrict__ dst,
                                 const float* __restrict__ ew,
                                 float* __restrict__ deg) {
    int e = blockIdx.x * blockDim.x + threadIdx.x;
    if (e < N_EDGES) atomicAdd(&deg[dst[e]], ew[e]);
}

__global__ void dinv_kernel(const float* __restrict__ deg, float* __restrict__ dinv) {
    int i = blockIdx.x * blockDim.x + threadIdx.x;
    if (i < N_NODES) dinv[i] = rsqrtf(deg[i]);   // deg >= 1 guaranteed by self-loop
}

__global__ void norm_kernel(const int* __restrict__ src, const int* __restrict__ dst,
                            const float* __restrict__ ew, const float* __restrict__ dinv,
                            float* __restrict__ norm) {
    int e = blockIdx.x * blockDim.x + threadIdx.x;
    if (e < N_EDGES) norm[e] = dinv[src[e]] * ew[e] * dinv[dst[e]];
}

__global__ void zero4_kernel(float4* __restrict__ p, int n4) {
    int i = blockIdx.x * blockDim.x + threadIdx.x;
    if (i < n4) p[i] = make_float4(0.f, 0.f, 0.f, 0.f);
}

// ---------------- WMMA f32 GEMM:  C[M x 64] = A[M x K] * B[K x 64] ----------------
// One wave per 16-row strip; 4 x (16x16) accumulator tiles cover all 64 columns.
// M must be a multiple of 16 (100000 = 6250*16), K a multiple of 4.
__global__ void __launch_bounds__(32)
gemm_wmma_f32(const float* __restrict__ A, const float* __restrict__ B,
              float* __restrict__ C, int K) {
    const int tile_m = blockIdx.x * 16;
    const int lane = threadIdx.x;       // 0..31
    const int hi   = lane >> 4;         // lane half: 0 -> K+0/1, 1 -> K+2/3
    const int lid  = lane & 15;

    const float* __restrict__ arow = A + (size_t)(tile_m + lid) * (size_t)K;

    v8f acc0 = {}, acc1 = {}, acc2 = {}, acc3 = {};

    for (int k = 0; k < K; k += 4) {
        // A fragment (16x4, 32-bit): lanes 0-15 carry K=k,k+1; lanes 16-31 K=k+2,k+3
        v2f a;
        a.x = arow[k + 2 * hi + 0];
        a.y = arow[k + 2 * hi + 1];

        // B fragment (4x16): row k+2*hi in .x, row k+2*hi+1 in .y, column = lid + 16*t
        const float* __restrict__ brow0 = B + (size_t)(k + 2 * hi + 0) * 64 + lid;
        const float* __restrict__ brow1 = B + (size_t)(k + 2 * hi + 1) * 64 + lid;

        v2f b;
        b.x = brow0[0];  b.y = brow1[0];
        acc0 = __builtin_amdgcn_wmma_f32_16x16x4_f32(false, a, false, b, (short)0, acc0, false, false);
        b.x = brow0[16]; b.y = brow1[16];
        acc1 = __builtin_amdgcn_wmma_f32_16x16x4_f32(false, a, false, b, (short)0, acc1, false, false);
        b.x = brow0[32]; b.y = brow1[32];
        acc2 = __builtin_amdgcn_wmma_f32_16x16x4_f32(false, a, false, b, (short)0, acc2, false, false);
        b.x = brow0[48]; b.y = brow1[48];
        acc3 = __builtin_amdgcn_wmma_f32_16x16x4_f32(false, a, false, b, (short)0, acc3, false, false);
    }

    // C/D layout: VGPR v holds row (v + 8*hi), column = lid (within 16-wide tile)
    float* __restrict__ cbase = C + (size_t)tile_m * 64;
    #pragma unroll
    for (int v = 0; v < 8; ++v) {
        const size_t roff = (size_t)(v + 8 * hi) * 64 + lid;
        cbase[roff +  0] = acc0[v];
        cbase[roff + 16] = acc1[v];
        cbase[roff + 32] = acc2[v];
        cbase[roff + 48] = acc3[v];
    }
}

// ---------------- edge aggregation:  agg[dst] += t[src] * norm ----------------
// 16 threads per edge, float4 (b128) gathers; atomics land in the 192MB L2.
__global__ void aggregate_kernel(const float4* __restrict__ t4,
                                 const int* __restrict__ src, const int* __restrict__ dst,
                                 const float* __restrict__ norm,
                                 float* __restrict__ agg) {
    size_t gid = (size_t)blockIdx.x * blockDim.x + threadIdx.x;
    size_t e = gid >> 4;
    int f4 = (int)(gid & 15);
    if (e >= (size_t)N_EDGES) return;
    int s = src[e], d = dst[e];
    float nrm = norm[e];
    float4 m = t4[(size_t)s * 16 + f4];
    float* dp = agg + (size_t)d * 64 + f4 * 4;
    atomicAdd(dp + 0, m.x * nrm);
    atomicAdd(dp + 1, m.y * nrm);
    atomicAdd(dp + 2, m.z * nrm);
    atomicAdd(dp + 3, m.w * nrm);
}

// h = leaky( agg + t*dinv^2 + bias )   (self-loop folded in; in-place on agg)
__global__ void epilogue_kernel(float4* __restrict__ agg4, const float4* __restrict__ t4,
                                const float* __restrict__ dinv, const float4* __restrict__ bias4) {
    size_t gid = (size_t)blockIdx.x * blockDim.x + threadIdx.x;
    if (gid >= (size_t)N_NODES * 16) return;
    size_t i = gid >> 4;
    int f4 = (int)(gid & 15);
    float di = dinv[i];
    float d2 = di * di;
    float4 a = agg4[gid], t = t4[gid], b = bias4[f4];
    float4 r;
    r.x = a.x + t.x * d2 + b.x;  r.x = r.x > 0.f ? r.x : NEG_SLOPE * r.x;
    r.y = a.y + t.y * d2 + b.y;  r.y = r.y > 0.f ? r.y : NEG_SLOPE * r.y;
    r.z = a.z + t.z * d2 + b.z;  r.z = r.z > 0.f ? r.z : NEG_SLOPE * r.z;
    r.w = a.w + t.w * d2 + b.w;  r.w = r.w > 0.f ? r.w : NEG_SLOPE * r.w;
    agg4[gid] = r;
}

__global__ void bias_leaky_kernel(float4* __restrict__ t4, const float4* __restrict__ bias4) {
    size_t gid = (size_t)blockIdx.x * blockDim.x + threadIdx.x;
    if (gid >= (size_t)N_NODES * 16) return;
    float4 t = t4[gid], b = bias4[gid & 15];
    float4 r;
    r.x = t.x + b.x;  r.x = r.x > 0.f ? r.x : NEG_SLOPE * r.x;
    r.y = t.y + b.y;  r.y = r.y > 0.f ? r.y : NEG_SLOPE * r.y;
    r.z = t.z + b.z;  r.z = r.z > 0.f ? r.z : NEG_SLOPE * r.z;
    r.w = t.w + b.w;  r.w = r.w > 0.f ? r.w : NEG_SLOPE * r.w;
    t4[gid] = r;
}

// logits = h @ Wm1 + bm1 ; softmax over 2 classes
__global__ void head_kernel(const float4* __restrict__ h4, const float* __restrict__ Wm1,
                            const float* __restrict__ bm1, float* __restrict__ out) {
    int i = blockIdx.x * blockDim.x + threadIdx.x;
    if (i >= N_NODES) return;
    float l0 = bm1[0], l1 = bm1[1];
    #pragma unroll
    for (int f4 = 0; f4 < 16; ++f4) {
        float4 v = h4[(size_t)i * 16 + f4];
        l0 += v.x * Wm1[8 * f4 + 0];  l1 += v.x * Wm1[8 * f4 + 1];
        l0 += v.y * Wm1[8 * f4 + 2];  l1 += v.y * Wm1[8 * f4 + 3];
        l0 += v.z * Wm1[8 * f4 + 4];  l1 += v.z * Wm1[8 * f4 + 5];
        l0 += v.w * Wm1[8 * f4 + 6];  l1 += v.w * Wm1[8 * f4 + 7];
    }
    float m = fmaxf(l0, l1);
    float e0 = __expf(l0 - m), e1 = __expf(l1 - m);
    float inv = 1.0f / (e0 + e1);
    out[2 * i + 0] = e0 * inv;
    out[2 * i + 1] = e1 * inv;
}

// ---------------- launch ----------------

extern "C" void kernel_launch(void* const* d_in, const int* in_sizes, int n_in,
                              void* d_out, int out_size, void* d_ws, size_t ws_size,
                              hipStream_t stream) {
    const float* X   = (const float*)d_in[0];
    const int*   ei  = (const int*)  d_in[1];   // [2, E] row-major: src row then dst row
    const float* ew  = (const float*)d_in[2];
    const float* W0  = (const float*)d_in[3];
    const float* b0  = (const float*)d_in[4];
    const float* W1  = (const float*)d_in[5];
    const float* b1  = (const float*)d_in[6];
    const float* Wm0 = (const float*)d_in[7];
    const float* bm0 = (const float*)d_in[8];
    const float* Wm1 = (const float*)d_in[9];
    const float* bm1 = (const float*)d_in[10];
    float* out = (float*)d_out;

    const int* src = ei;
    const int* dst = ei + N_EDGES;

    // workspace carve-up
    char* w = (char*)d_ws;
    float* deg  = (float*)w;                w += (size_t)N_NODES * 4;
    float* dinv = (float*)w;                w += (size_t)N_NODES * 4;
    float* norm = (float*)w;                w += (size_t)N_EDGES * 4;
    float* bufT = (float*)w;                w += (size_t)N_NODES * 64 * 4;   // linear outputs
    float* bufA = (float*)w;                w += (size_t)N_NODES * 64 * 4;   // agg / activations

    const int TB = 256;
    const int nodeBlocks = (N_NODES + TB - 1) / TB;
    const int edgeBlocks = (N_EDGES + TB - 1) / TB;
    const int feat4Elems = N_NODES * 16;                       // float4 elements
    const int feat4Blocks = (feat4Elems + TB - 1) / TB;
    const size_t aggThreads = (size_t)N_EDGES * 16;            // 16 threads/edge
    const int aggBlocks = (int)((aggThreads + TB - 1) / TB);
    const int gemmBlocks = N_NODES / 16;                       // 6250, exact

    // normalization coefficients (shared by both conv layers)
    init_deg_kernel<<<nodeBlocks, TB, 0, stream>>>(deg);
    deg_accum_kernel<<<edgeBlocks, TB, 0, stream>>>(dst, ew, deg);
    dinv_kernel<<<nodeBlocks, TB, 0, stream>>>(deg, dinv);
    norm_kernel<<<edgeBlocks, TB, 0, stream>>>(src, dst, ew, dinv, norm);

    // layer 1: t = X @ W0 ; agg = scatter(t*norm) ; h1 = leaky(agg + t*dinv^2 + b0)
    gemm_wmma_f32<<<gemmBlocks, 32, 0, stream>>>(X, W0, bufT, IN_F);
    zero4_kernel<<<feat4Blocks, TB, 0, stream>>>((float4*)bufA, feat4Elems);
    aggregate_kernel<<<aggBlocks, TB, 0, stream>>>((const float4*)bufT, src, dst, norm, bufA);
    epilogue_kernel<<<feat4Blocks, TB, 0, stream>>>((float4*)bufA, (const float4*)bufT, dinv, (const float4*)b0);

    // layer 2: t = h1 @ W1 ; same aggregation ; h2 in bufA
    gemm_wmma_f32<<<gemmBlocks, 32, 0, stream>>>(bufA, W1, bufT, HID);
    zero4_kernel<<<feat4Blocks, TB, 0, stream>>>((float4*)bufA, feat4Elems);
    aggregate_kernel<<<aggBlocks, TB, 0, stream>>>((const float4*)bufT, src, dst, norm, bufA);
    epilogue_kernel<<<feat4Blocks, TB, 0, stream>>>((float4*)bufA, (const float4*)bufT, dinv, (const float4*)b1);

    // MLP head: t = leaky(h2 @ Wm0 + bm0) ; out = softmax(t @ Wm1 + bm1)
    gemm_wmma_f32<<<gemmBlocks, 32, 0, stream>>>(bufA, Wm0, bufT, HID);
    bias_leaky_kernel<<<feat4Blocks, TB, 0, stream>>>((float4*)bufT, (const float4*)bm0);
    head_kernel<<<nodeBlocks, TB, 0, stream>>>((const float4*)bufT, Wm1, bm1, out);
}